// CF_87866440942102
// MI455X (gfx1250) — compile-verified
//
#include <hip/hip_runtime.h>
#include <math.h>

typedef __attribute__((ext_vector_type(2))) float v2f;
typedef __attribute__((ext_vector_type(8))) float v8f;

// ---------------------------------------------------------------------------
// Kernel 1: scalars (std, kl_global)
// ---------------------------------------------------------------------------
__global__ void cf_scalars_kernel(const float* __restrict__ alpha,
                                  const float* __restrict__ mgb,
                                  const float* __restrict__ sgb,
                                  const float* __restrict__ mgbp,
                                  const float* __restrict__ sgbp,
                                  float* __restrict__ out_std,
                                  float* __restrict__ out_klg) {
    if (threadIdx.x == 0 && blockIdx.x == 0) {
        out_std[0] = sqrtf(1.0f / fabsf(alpha[0]));
        float mq = mgb[0];
        float sq = fabsf(sgb[0]);
        float mp = mgbp[0];
        float sp = fabsf(sgbp[0]);
        float d  = mq - mp;
        out_klg[0] = logf(sp) - logf(sq) + (sq * sq + d * d) / (2.0f * sp * sp) - 0.5f;
    }
}

// ---------------------------------------------------------------------------
// Kernel 2: kl_bias (U,) + kl_entity (U,32).  One wave32 per gathered row:
// lane e handles entity component e.  Reads are two coalesced 128B lines per
// row (mean half, scale half); the kl_entity store is one coalesced 128B line.
// ---------------------------------------------------------------------------
__global__ void cf_kl_kernel(const int*   __restrict__ xu0,
                             const int*   __restrict__ xu1,
                             const float* __restrict__ bias_params,   // (N+M, 2)
                             const float* __restrict__ ent_params,    // (N+M, 64)
                             const float* __restrict__ mgbp,          // (2,)
                             const float* __restrict__ sgbp,          // (2,)
                             const float* __restrict__ mgep,          // (2,32)
                             const float* __restrict__ sgep,          // (2,32)
                             float* __restrict__ out_kb,              // (U,)
                             float* __restrict__ out_ke,              // (U,32)
                             int U1, int U) {
    const int lane = threadIdx.x & 31;
    const int wave = threadIdx.x >> 5;
    const int row  = blockIdx.x * (blockDim.x >> 5) + wave;
    if (row >= U) return;

    const int g   = (row < U1) ? 0 : 1;
    const int idx = (row < U1) ? xu0[row] : xu1[row - U1];

    const float* ep = ent_params + (long long)idx * 64;
    float mq = ep[lane];
    float sq = fabsf(ep[32 + lane]);
    float mp = mgep[g * 32 + lane];
    float sp = fabsf(sgep[g * 32 + lane]);
    float d  = mq - mp;
    out_ke[(long long)row * 32 + lane] =
        logf(sp) - logf(sq) + (sq * sq + d * d) / (2.0f * sp * sp) - 0.5f;

    if (lane == 0) {
        float bm  = bias_params[(long long)idx * 2];
        float bs  = fabsf(bias_params[(long long)idx * 2 + 1]);
        float bmp = mgbp[g];
        float bsp = fabsf(sgbp[g]);
        float bd  = bm - bmp;
        out_kb[row] =
            logf(bsp) - logf(bs) + (bs * bs + bd * bd) / (2.0f * bsp * bsp) - 0.5f;
    }
}

// ---------------------------------------------------------------------------
// Kernel 3: pred via V_WMMA_F32_16X16X4_F32.
// One wave handles 16 pairs.  A(16x4 f32) and B(4x16 f32) share the same lane
// layout: lane r (0..15) carries K={4j,4j+1} of pair r's vectors, lane r+16
// carries K={4j+2,4j+3}.  8 chained WMMAs accumulate the full K=32 reduction
// into a 16x16 f32 accumulator whose diagonal holds the 16 dot products.
// Diagonal D[i,i]: (VGPR i, lane i) for i<8, (VGPR i-8, lane i+16) for i>=8 —
// extracted with ds_bpermute broadcasts (in-wave, no barrier).
// EXEC is all-ones throughout (WMMA requirement): loads are clamped, only the
// final store is predicated.
// ---------------------------------------------------------------------------
__global__ void cf_pred_wmma_kernel(const int*   __restrict__ x0,
                                    const int*   __restrict__ x1,
                                    const int*   __restrict__ xu0,
                                    const int*   __restrict__ xu1,
                                    const float* __restrict__ mgb,
                                    const float* __restrict__ bias_params, // (N+M,2)
                                    const float* __restrict__ ent_params,  // (N+M,64)
                                    float* __restrict__ out_pred,          // (B,)
                                    int B) {
    const int lane = threadIdx.x & 31;
    const int wave = threadIdx.x >> 5;
    const int r    = lane & 15;   // pair within tile
    const int h    = lane >> 4;   // K-half select

    const long long tile = (long long)blockIdx.x * (blockDim.x >> 5) + wave;
    const long long p    = tile * 16 + r;
    const long long pc   = (p < B) ? p : (long long)(B - 1);   // clamp, keep EXEC full

    const int ia = xu0[x0[pc]];
    const int ib = xu1[x1[pc]];
    const float* ap = ent_params + (long long)ia * 64;   // mean half = first 32 floats
    const float* bp = ent_params + (long long)ib * 64;

    v8f c = {0.f, 0.f, 0.f, 0.f, 0.f, 0.f, 0.f, 0.f};
#pragma unroll
    for (int j = 0; j < 8; ++j) {
        v2f a = *(const v2f*)(ap + 4 * j + 2 * h);   // 8B-aligned global_load_b64
        v2f b = *(const v2f*)(bp + 4 * j + 2 * h);
        c = __builtin_amdgcn_wmma_f32_16x16x4_f32(
                /*neg_a=*/false, a, /*neg_b=*/false, b,
                /*c_mod=*/(short)0, c, /*reuse_a=*/false, /*reuse_b=*/false);
    }

    // Diagonal extraction: lane r needs accumulator VGPR kneed from lane src.
    const int src   = (r < 8) ? r : (r + 8);
    const int kneed = (r < 8) ? r : (r - 8);
    float dot = 0.0f;
#pragma unroll
    for (int k = 0; k < 8; ++k) {
        int t = __builtin_amdgcn_ds_bpermute(src << 2, __float_as_int(c[k]));
        if (k == kneed) dot = __int_as_float(t);
    }

    if (h == 0 && p < B) {
        out_pred[p] = mgb[0]
                    + bias_params[(long long)ia * 2]
                    + bias_params[(long long)ib * 2]
                    + dot;
    }
}

// ---------------------------------------------------------------------------
// Launch
// ---------------------------------------------------------------------------
extern "C" void kernel_launch(void* const* d_in, const int* in_sizes, int n_in,
                              void* d_out, int out_size, void* d_ws, size_t ws_size,
                              hipStream_t stream) {
    const int*   x0    = (const int*)  d_in[0];
    const int*   x1    = (const int*)  d_in[1];
    const int*   xu0   = (const int*)  d_in[2];
    const int*   xu1   = (const int*)  d_in[3];
    const float* alpha = (const float*)d_in[4];
    const float* mgb   = (const float*)d_in[5];   // mean_global_bias
    const float* sgb   = (const float*)d_in[6];   // scale_global_bias
    const float* mgbp  = (const float*)d_in[7];   // mean_global_bias_prior
    const float* sgbp  = (const float*)d_in[8];   // scale_global_bias_prior
    const float* mgrbp = (const float*)d_in[9];   // mean_group_bias_prior (2,)
    const float* sgrbp = (const float*)d_in[10];  // scale_group_bias_prior (2,)
    const float* mgep  = (const float*)d_in[11];  // mean_group_entity_prior (2,32)
    const float* sgep  = (const float*)d_in[12];  // scale_group_entity_prior (2,32)
    const float* biasp = (const float*)d_in[13];  // bias_params (N+M,2)
    const float* entp  = (const float*)d_in[14];  // entity_params (N+M,64)

    const int B  = in_sizes[0];
    const int U1 = in_sizes[2];
    const int U2 = in_sizes[3];
    const int U  = U1 + U2;

    float* out      = (float*)d_out;
    float* out_pred = out;                 // (B,)
    float* out_std  = out + B;             // (1,)
    float* out_klg  = out + B + 1;         // (1,)
    float* out_kb   = out + B + 2;         // (U,)
    float* out_ke   = out + B + 2 + U;     // (U,32)

    cf_scalars_kernel<<<1, 32, 0, stream>>>(alpha, mgb, sgb, mgbp, sgbp,
                                            out_std, out_klg);

    // kl: 8 waves/block, one row per wave
    const int klBlocks = (U + 7) / 8;
    cf_kl_kernel<<<klBlocks, 256, 0, stream>>>(xu0, xu1, biasp, entp,
                                               mgrbp, sgrbp, mgep, sgep,
                                               out_kb, out_ke, U1, U);

    // pred: 8 waves/block * 16 pairs/wave = 128 pairs/block
    const int predBlocks = (B + 127) / 128;
    cf_pred_wmma_kernel<<<predBlocks, 256, 0, stream>>>(x0, x1, xu0, xu1,
                                                        mgb, biasp, entp,
                                                        out_pred, B);
}